// HungarianPointMatcher_29437705847328
// MI455X (gfx1250) — compile-verified
//
#include <hip/hip_runtime.h>
#include <math.h>

typedef __attribute__((ext_vector_type(2))) float v2f;
typedef __attribute__((ext_vector_type(8))) float v8f;

#define N_PTS  1024   // N == M == 1024
#define P_PTS  20
#define LAPN   1024

// ---------------------------------------------------------------------------
// Phase 1: cost[i][j] = (1/P) * sum_p sqrt(|pred[i,p] - gt[j,p]|^2)
// Per point p, the 16x16 tile of squared distances is one K=4 f32 WMMA:
//   A'[i] = [ax, ay, |a|^2, 1]   (16x4)
//   B'[j] = [-2bx, -2by, 1, |b|^2] (4x16)
//   A' x B' = |a|^2 + |b|^2 - 2 a.b = d^2
// A-layout (16x4 f32): lanes 0-15 hold K=0,1 for M=lane; lanes 16-31 hold K=2,3.
// B-layout (4x16 f32): lanes 0-15 hold K=0,1 for N=lane; lanes 16-31 hold K=2,3.
// D-layout (16x16 f32): VGPR e holds M=e (lanes 0-15) / M=e+8 (lanes 16-31),
// N = lane & 15.
// ---------------------------------------------------------------------------
__global__ __launch_bounds__(32)
void cost_tile_kernel(const float* __restrict__ pred,
                      const float* __restrict__ gt,
                      float* __restrict__ cost)
{
    const int lane = threadIdx.x;          // 0..31, one wave per 16x16 tile
    const int ti   = blockIdx.x;           // row tile (0..63)
    const int tj   = blockIdx.y;           // col tile (0..63)
    const bool hi  = lane >= 16;
    const int r    = ti * 16 + (lane & 15);
    const int c    = tj * 16 + (lane & 15);

    v8f acc = {0.f, 0.f, 0.f, 0.f, 0.f, 0.f, 0.f, 0.f};

#pragma unroll
    for (int p = 0; p < P_PTS; ++p) {
        const float2 ap = *(const float2*)(pred + (r * P_PTS + p) * 2);
        const float2 bp = *(const float2*)(gt   + (c * P_PTS + p) * 2);

        v2f A, B;
        if (!hi) {
            A.x = ap.x;            A.y = ap.y;            // K=0,1 : ax, ay
            B.x = -2.0f * bp.x;    B.y = -2.0f * bp.y;    // K=0,1 : -2bx, -2by
        } else {
            A.x = ap.x * ap.x + ap.y * ap.y;  A.y = 1.0f; // K=2,3 : |a|^2, 1
            B.x = 1.0f;  B.y = bp.x * bp.x + bp.y * bp.y; // K=2,3 : 1, |b|^2
        }

        v8f zero = {0.f, 0.f, 0.f, 0.f, 0.f, 0.f, 0.f, 0.f};
        // D = A' x B' + 0  ->  16x16 tile of squared distances for point p
        v8f d2 = __builtin_amdgcn_wmma_f32_16x16x4_f32(
            /*neg_a=*/false, A, /*neg_b=*/false, B,
            /*c_mod=*/(short)0, zero, /*reuse_a=*/false, /*reuse_b=*/false);

#pragma unroll
        for (int e = 0; e < 8; ++e)
            acc[e] += __builtin_amdgcn_sqrtf(fmaxf(d2[e], 0.0f)); // v_sqrt_f32
    }

#pragma unroll
    for (int e = 0; e < 8; ++e) {
        const int row = ti * 16 + e + (hi ? 8 : 0);
        cost[row * N_PTS + c] = acc[e] * (1.0f / (float)P_PTS);
    }
}

// ---------------------------------------------------------------------------
// Phase 2: Jonker-Volgenant shortest-augmenting-path LAP, one workgroup of
// 1024 threads (32 wave32 waves). Thread j owns column j. All solver state
// in LDS; cost rows stream from L2 (4 MB matrix fits easily in 192 MB L2).
// ---------------------------------------------------------------------------
__device__ __forceinline__ unsigned fkey(float f) {
    unsigned u = __float_as_uint(f);
    // monotone (order-preserving) mapping float -> uint
    return (u & 0x80000000u) ? ~u : (u | 0x80000000u);
}

__global__ __launch_bounds__(1024)
void lap_kernel(const float* __restrict__ cost, int* __restrict__ col4row_out)
{
    __shared__ float u[LAPN], v[LAPN], shortest[LAPN];
    __shared__ int   path[LAPN], col4row[LAPN], row4col[LAPN];
    __shared__ unsigned char SC[LAPN], SR[LAPN];
    __shared__ unsigned long long red[LAPN];
    __shared__ int   s_i, s_sink;
    __shared__ float s_minval;

    const int tid = threadIdx.x;

    u[tid] = 0.0f; v[tid] = 0.0f;
    col4row[tid] = -1; row4col[tid] = -1;
    __syncthreads();

    for (int cur = 0; cur < LAPN; ++cur) {
        shortest[tid] = __builtin_inff();
        path[tid] = -1; SC[tid] = 0; SR[tid] = 0;
        if (tid == 0) { s_i = cur; s_minval = 0.0f; s_sink = -1; }
        __syncthreads();

        // Dijkstra-like shortest augmenting path
        while (true) {
            const int   i  = s_i;
            const float mv = s_minval;
            if (tid == 0) SR[i] = 1;

            // parallel relaxation over all unscanned columns
            if (!SC[tid]) {
                const float d = mv + cost[i * LAPN + tid] - u[i] - v[tid];
                if (d < shortest[tid]) { shortest[tid] = d; path[tid] = i; }
            }
            __syncthreads();

            // argmin over !SC columns; (value, index) lexicographic min
            // reproduces np.argmin's first-occurrence tie-break.
            unsigned long long key = ~0ull;
            if (!SC[tid])
                key = (((unsigned long long)fkey(shortest[tid])) << 32) |
                      (unsigned long long)(unsigned)tid;
            red[tid] = key;
            __syncthreads();
#pragma unroll
            for (int s = 512; s > 0; s >>= 1) {
                if (tid < s) {
                    const unsigned long long o = red[tid + s];
                    if (o < red[tid]) red[tid] = o;
                }
                __syncthreads();
            }

            if (tid == 0) {
                const int k = (int)(red[0] & 0xFFFFFFFFull);
                s_minval = shortest[k];
                SC[k] = 1;
                const int rk = row4col[k];
                if (rk < 0) s_sink = k;
                else        s_i    = rk;
            }
            __syncthreads();
            if (s_sink >= 0) break;
        }

        // dual variable update (parallel over rows / cols)
        const float mv = s_minval;
        if (tid == cur)      u[tid] += mv;
        else if (SR[tid])    u[tid] += mv - shortest[col4row[tid]];
        if (SC[tid])         v[tid] -= mv - shortest[tid];
        __syncthreads();

        // augment along the path (serial, thread 0)
        if (tid == 0) {
            int j = s_sink;
            while (true) {
                const int i2 = path[j];
                row4col[j] = i2;
                const int t = col4row[i2];
                col4row[i2] = j;
                j = t;
                if (i2 == cur) break;
            }
        }
        __syncthreads();
    }

    col4row_out[tid] = col4row[tid];
}

// ---------------------------------------------------------------------------
// Phase 3: matched[i] = gt[col4row[i]]; conf[i] = exp(-cost[i, col4row[i]])
// d_out = matched (1024*20*2 floats) || conf (1024 floats)
// ---------------------------------------------------------------------------
__global__ __launch_bounds__(256)
void gather_kernel(const float* __restrict__ gt,
                   const float* __restrict__ cost,
                   const int* __restrict__ col4row,
                   float* __restrict__ out)
{
    const int i = blockIdx.x * blockDim.x + threadIdx.x;
    if (i >= N_PTS) return;
    const int c = col4row[i];

    const float4* src = (const float4*)(gt + c * (P_PTS * 2));
    float4* dst       = (float4*)(out + i * (P_PTS * 2));
#pragma unroll
    for (int k = 0; k < (P_PTS * 2) / 4; ++k) dst[k] = src[k];

    // exp(-x) = exp2(-x * log2(e)) via native v_exp_f32
    const float x = cost[i * N_PTS + c];
    out[N_PTS * P_PTS * 2 + i] =
        __builtin_amdgcn_exp2f(-x * 1.44269504088896340736f);
}

// ---------------------------------------------------------------------------
extern "C" void kernel_launch(void* const* d_in, const int* in_sizes, int n_in,
                              void* d_out, int out_size, void* d_ws, size_t ws_size,
                              hipStream_t stream)
{
    const float* pred = (const float*)d_in[0];   // (1024, 20, 2) f32
    const float* gt   = (const float*)d_in[1];   // (1024, 20, 2) f32

    float* cost = (float*)d_ws;                                  // 4 MB
    int*   c4r  = (int*)((char*)d_ws + (size_t)N_PTS * N_PTS * sizeof(float));

    dim3 grid_cost(N_PTS / 16, N_PTS / 16);      // 64 x 64 tiles, 1 wave each
    cost_tile_kernel<<<grid_cost, 32, 0, stream>>>(pred, gt, cost);

    lap_kernel<<<1, 1024, 0, stream>>>(cost, c4r);

    gather_kernel<<<(N_PTS + 255) / 256, 256, 0, stream>>>(gt, cost, c4r,
                                                           (float*)d_out);
}